// GNLoss_31825707663476
// MI455X (gfx1250) — compile-verified
//
#include <hip/hip_runtime.h>
#include <math.h>

typedef __attribute__((ext_vector_type(2))) float v2f;
typedef __attribute__((ext_vector_type(8))) float v8f;

// Problem dims (fixed by the reference setup_inputs()).
#define BB 4
#define CC 128
#define HH 240
#define WW 320
#define NN 4000
#define BN (BB * NN)          // 16000 points
#define NWAVES (BN / 4)       // 4000 waves, 4 points each
#define WAVES_PER_BLOCK 8
#define NBLOCKS (NWAVES / WAVES_PER_BLOCK)  // 500

// Per wave: 4 points. Rows of M (16 x 128): row = 4*p + v,
//   v=0: f_s (bilinear sample), v=1: f_t (target row),
//   v=2: Jx (bilinear of d/dx),  v=3: Jy (bilinear of d/dy).
// Gram G = M M^T accumulated with v_wmma_f32_16x16x4_f32 over 32 K-chunks.
// f32 WMMA A (16x4) and B(=A^T, 4x16) have identical per-lane register
// contents, so a single v2f feeds both operands.
__global__ __launch_bounds__(256) void gn_gram_kernel(
    const float* __restrict__ fb, const float* __restrict__ f_t,
    const float* __restrict__ ub, const float* __restrict__ noise,
    float* __restrict__ ws_e1, float* __restrict__ ws_ld) {
  const int lane  = threadIdx.x & 31;
  const int wslot = threadIdx.x >> 5;
  const int gwave = blockIdx.x * WAVES_PER_BLOCK + wslot;
  const int p     = (lane & 15) >> 2;  // point within wave (0..3)
  const int v     = lane & 3;          // vector type (0..3)
  const int khalf = lane >> 4;         // K sub-pair (0 or 1)

  const int n = gwave * 4 + p;         // global point index
  const int b = n / NN;

  // Sample location
  const float sx = ub[2 * n + 0] + noise[2 * n + 0];
  const float sy = ub[2 * n + 1] + noise[2 * n + 1];
  int x0 = (int)floorf(sx); x0 = min(max(x0, 0), WW - 2);
  int y0 = (int)floorf(sy); y0 = min(max(y0, 0), HH - 2);
  const float wx = fminf(fmaxf(sx - (float)x0, 0.0f), 1.0f);
  const float wy = fminf(fmaxf(sy - (float)y0, 0.0f), 1.0f);
  const float w00 = (1.0f - wx) * (1.0f - wy);
  const float w01 = wx * (1.0f - wy);
  const float w10 = (1.0f - wx) * wy;
  const float w11 = wx * wy;

  // Uniform gather description: val(c) = sum_j wts[j] * base[c*cstride + offs[j]]
  const float* base;
  size_t cstride;
  int offs[8];
  float wts[8];
  if (v == 1) {
    base = f_t + (size_t)n * CC;
    cstride = 1;
    #pragma unroll
    for (int j = 0; j < 8; ++j) { offs[j] = 0; wts[j] = 0.0f; }
    wts[0] = 1.0f;
  } else {
    base = fb + (size_t)b * CC * HH * WW;
    cstride = (size_t)HH * WW;
    if (v == 0) {
      const int r0 = y0 * WW + x0, r1 = r0 + WW;
      offs[0] = r0;     offs[1] = r0 + 1; offs[2] = r1;     offs[3] = r1 + 1;
      offs[4] = r0;     offs[5] = r0;     offs[6] = r0;     offs[7] = r0;
      wts[0] = w00; wts[1] = w01; wts[2] = w10; wts[3] = w11;
      wts[4] = 0.0f; wts[5] = 0.0f; wts[6] = 0.0f; wts[7] = 0.0f;
    } else if (v == 2) {
      // Jx: gx(y,x) = 0.5*(P(y,min(x+1,W-1)) - P(y,max(x-1,0)))
      const int xA = max(x0 - 1, 0), xB = x0 + 1, xC = min(x0 + 2, WW - 1);
      const int rA = y0 * WW, rB = (y0 + 1) * WW;
      offs[0] = rA + xB; offs[1] = rA + xA; offs[2] = rA + xC; offs[3] = rA + x0;
      offs[4] = rB + xB; offs[5] = rB + xA; offs[6] = rB + xC; offs[7] = rB + x0;
      wts[0] =  0.5f * w00; wts[1] = -0.5f * w00;
      wts[2] =  0.5f * w01; wts[3] = -0.5f * w01;
      wts[4] =  0.5f * w10; wts[5] = -0.5f * w10;
      wts[6] =  0.5f * w11; wts[7] = -0.5f * w11;
    } else {
      // Jy: gy(y,x) = 0.5*(P(min(y+1,H-1),x) - P(max(y-1,0),x))
      const int yA = max(y0 - 1, 0), yB = y0 + 1, yC = min(y0 + 2, HH - 1);
      const int x1 = x0 + 1;
      offs[0] = yB * WW + x0; offs[1] = yA * WW + x0;
      offs[2] = yB * WW + x1; offs[3] = yA * WW + x1;
      offs[4] = yC * WW + x0; offs[5] = y0 * WW + x0;
      offs[6] = yC * WW + x1; offs[7] = y0 * WW + x1;
      wts[0] =  0.5f * w00; wts[1] = -0.5f * w00;
      wts[2] =  0.5f * w01; wts[3] = -0.5f * w01;
      wts[4] =  0.5f * w10; wts[5] = -0.5f * w10;
      wts[6] =  0.5f * w11; wts[7] = -0.5f * w11;
    }
  }

  // Accumulate Gram over K-chunks of 4 channels.
  v8f acc = {0.f, 0.f, 0.f, 0.f, 0.f, 0.f, 0.f, 0.f};
  for (int kc = 0; kc < CC / 4; ++kc) {
    const int c0 = 4 * kc + 2 * khalf;
    const float* b0p = base + (size_t)c0 * cstride;
    const float* b1p = b0p + cstride;
    float m0 = 0.0f, m1 = 0.0f;
    #pragma unroll
    for (int j = 0; j < 8; ++j) {
      m0 = fmaf(wts[j], b0p[offs[j]], m0);
      m1 = fmaf(wts[j], b1p[offs[j]], m1);
    }
    v2f a; a.x = m0; a.y = m1;
    acc = __builtin_amdgcn_wmma_f32_16x16x4_f32(
        false, a, false, a, (short)0, acc, false, false);
  }

  // Stage Gram (16x16 f32 C/D layout) into LDS:
  // G[r][col]   = Gs[w][r][col]        (rows 0..7,  lanes 0..15 hold cols)
  // G[r+8][col] = Gs[w][r][col+16]     (rows 8..15, lanes 16..31 hold cols)
  __shared__ float Gs[WAVES_PER_BLOCK][8][32];
  #pragma unroll
  for (int r = 0; r < 8; ++r) Gs[wslot][r][lane] = acc[r];
  __syncthreads();

  float e1p = 0.0f, ldp = 0.0f;
  if (lane < 4) {
    const int t = lane;                 // this lane finalizes point t
    const int n2 = gwave * 4 + t;
    auto G = [&](int i, int j) -> float {
      const int R = 4 * t + i, Cj = 4 * t + j;
      return (R < 8) ? Gs[wslot][R][Cj] : Gs[wslot][R - 8][Cj + 16];
    };
    const float Sss = G(0, 0), Stt = G(1, 1);
    const float Jxs = G(2, 0), Jxt = G(2, 1);
    const float Jys = G(3, 0), Jyt = G(3, 1);
    const float Jxx = G(2, 2), Jxy = G(2, 3), Jyy = G(3, 3);

    const float ins = 1.0f / (sqrtf(Sss) + 1e-12f);
    const float itn = 1.0f / (sqrtf(Stt) + 1e-12f);
    const float b0v = Jxs * ins - Jxt * itn;   // Jx . r
    const float b1v = Jys * ins - Jyt * itn;   // Jy . r
    const float H00 = Jxx + 1e-9f, H11 = Jyy + 1e-9f, H01 = Jxy;
    const float det = H00 * H11 - H01 * H01;
    const float hb0 = (H11 * b0v - H01 * b1v) / det;  // (H^-1 b)_x
    const float hb1 = (H00 * b1v - H01 * b0v) / det;  // (H^-1 b)_y
    // d = ub - miu = -noise + H^-1 b
    const float d0 = hb0 - noise[2 * n2 + 0];
    const float d1 = hb1 - noise[2 * n2 + 1];
    e1p = 0.5f * (d0 * (H00 * d0 + H01 * d1) + d1 * (H01 * d0 + H11 * d1));
    ldp = logf(det);
  }
  // Sum the 4 finalize lanes (others hold 0) down to lane 0.
  e1p += __shfl_down(e1p, 2, 32);
  ldp += __shfl_down(ldp, 2, 32);
  e1p += __shfl_down(e1p, 1, 32);
  ldp += __shfl_down(ldp, 1, 32);
  if (lane == 0) {
    ws_e1[gwave] = e1p;
    ws_ld[gwave] = ldp;
  }
}

__global__ __launch_bounds__(256) void gn_reduce_kernel(
    const float* __restrict__ ws_e1, const float* __restrict__ ws_ld,
    float* __restrict__ out) {
  __shared__ float s1[256], s2[256];
  float a = 0.0f, l = 0.0f;
  for (int i = threadIdx.x; i < NWAVES; i += 256) {
    a += ws_e1[i];
    l += ws_ld[i];
  }
  s1[threadIdx.x] = a;
  s2[threadIdx.x] = l;
  __syncthreads();
  for (int s = 128; s > 0; s >>= 1) {
    if (threadIdx.x < (unsigned)s) {
      s1[threadIdx.x] += s1[threadIdx.x + s];
      s2[threadIdx.x] += s2[threadIdx.x + s];
    }
    __syncthreads();
  }
  if (threadIdx.x == 0) {
    const float e1 = s1[0];
    const float e2 = (float)BN * 1.8378770664093453f - 0.5f * s2[0];
    const float e = 1.0f * e1 + (2.0f / 7.0f) * e2;
    out[0] = e;
    out[1] = e1;
    out[2] = e2;
  }
}

extern "C" void kernel_launch(void* const* d_in, const int* in_sizes, int n_in,
                              void* d_out, int out_size, void* d_ws, size_t ws_size,
                              hipStream_t stream) {
  const float* fb    = (const float*)d_in[0];  // (B,C,H,W)
  const float* f_t   = (const float*)d_in[1];  // (B*N,C)
  const float* ub    = (const float*)d_in[2];  // (B,N,2)
  const float* noise = (const float*)d_in[3];  // (B,N,2)
  float* out = (float*)d_out;                  // (e, e1, e2)
  float* ws  = (float*)d_ws;
  float* ws_e1 = ws;
  float* ws_ld = ws + NWAVES;

  gn_gram_kernel<<<NBLOCKS, 256, 0, stream>>>(fb, f_t, ub, noise, ws_e1, ws_ld);
  gn_reduce_kernel<<<1, 256, 0, stream>>>(ws_e1, ws_ld, out);
}